// DeformAttn_26207890440752
// MI455X (gfx1250) — compile-verified
//
#include <hip/hip_runtime.h>

typedef __attribute__((ext_vector_type(2))) float v2f;
typedef __attribute__((ext_vector_type(8))) float v8f;

#define B_    2
#define D_    256
#define NH_   8
#define NL_   4
#define NP_   4
#define HD_   32
#define NQ_   21760
#define NK_   21760
#define MROWS (B_ * NQ_)   /* 43520 rows for every GEMM (B*NQ == B*NK) */

// ---------------------------------------------------------------------------
// GEMM: C[M,N] = A[M,256] * W[256,N] + bias[N], fp32 via V_WMMA_F32_16X16X4_F32.
// One wave32 computes a 64x16 C tile (4 sub-tiles of 16x16) so the strided
// B fragment is loaded ONCE per K-step and reused by 4 WMMAs:
//   per K-step: 4x contiguous b64 A loads + 2x b32 B loads -> 4x wmma.
//   A frag (16x4 f32):  lane L holds row M=L%16, K = 2*(L/16)+{0,1}
//   B frag (4x16 f32):  lane L holds col N=L%16, K = 2*(L/16)+{0,1}
//   C/D  (16x16 f32):   lane L holds col N=L%16, VGPR r = row M = 8*(L/16)+r
// 8 waves per block. M must be a multiple of 64 (43520 = 680*64).
// ---------------------------------------------------------------------------
__global__ __launch_bounds__(256)
void gemm256_wmma_f32(const float* __restrict__ A, const float* __restrict__ W,
                      const float* __restrict__ bias, float* __restrict__ C,
                      int M, int N)
{
    const int lane = threadIdx.x & 31;
    const int wave = threadIdx.x >> 5;
    const int ntN  = N >> 4;
    const long tile = (long)blockIdx.x * 8 + wave;     // 64-row tiles
    const long ntot = (long)(M >> 6) * ntN;
    if (tile >= ntot) return;                 // wave-uniform: EXEC stays all-ones
    const int tileM = (int)(tile / ntN);      // in units of 64 rows
    const int tileN = (int)(tile % ntN);      // in units of 16 cols
    const int half  = lane >> 4;              // 0 or 1 (K sub-slice)
    const int mr    = lane & 15;              // A-row / B-col / C-col

    const float* arow0 = A + (long)(tileM * 64 + mr) * 256;
    const float* arow1 = arow0 + 16 * 256;
    const float* arow2 = arow0 + 32 * 256;
    const float* arow3 = arow0 + 48 * 256;
    const float* wcol  = W + tileN * 16 + mr;

    v8f c0 = {0.f,0.f,0.f,0.f,0.f,0.f,0.f,0.f};
    v8f c1 = c0, c2 = c0, c3 = c0;

#pragma unroll 4
    for (int k = 0; k < 256; k += 4) {
        const int ka = k + 2 * half;
        v2f b;
        b.x = wcol[(long)ka * N];
        b.y = wcol[(long)(ka + 1) * N];
        v2f a0, a1, a2, a3;
        a0.x = arow0[ka]; a0.y = arow0[ka + 1];
        a1.x = arow1[ka]; a1.y = arow1[ka + 1];
        a2.x = arow2[ka]; a2.y = arow2[ka + 1];
        a3.x = arow3[ka]; a3.y = arow3[ka + 1];
        c0 = __builtin_amdgcn_wmma_f32_16x16x4_f32(false, a0, false, b, (short)0, c0, false, false);
        c1 = __builtin_amdgcn_wmma_f32_16x16x4_f32(false, a1, false, b, (short)0, c1, false, false);
        c2 = __builtin_amdgcn_wmma_f32_16x16x4_f32(false, a2, false, b, (short)0, c2, false, false);
        c3 = __builtin_amdgcn_wmma_f32_16x16x4_f32(false, a3, false, b, (short)0, c3, false, false);
    }

    const float bb = bias[tileN * 16 + mr];
    const long  ncol = tileN * 16 + mr;
    float* crow0 = C + (long)(tileM * 64 +  0 + 8 * half) * N + ncol;
    float* crow1 = C + (long)(tileM * 64 + 16 + 8 * half) * N + ncol;
    float* crow2 = C + (long)(tileM * 64 + 32 + 8 * half) * N + ncol;
    float* crow3 = C + (long)(tileM * 64 + 48 + 8 * half) * N + ncol;
#pragma unroll
    for (int r = 0; r < 8; ++r) {
        crow0[(long)r * N] = c0[r] + bb;
        crow1[(long)r * N] = c1[r] + bb;
        crow2[(long)r * N] = c2[r] + bb;
        crow3[(long)r * N] = c3[r] + bb;
    }
}

// ---------------------------------------------------------------------------
// In-place epilogue on d_out: softmax over 16 attn logits per (b,q,h), and
// loc = ref + off / (W_l, H_l) where the off logits already live in the loc slot.
// One thread per (b,q,h).
// ---------------------------------------------------------------------------
__global__ __launch_bounds__(256)
void loc_softmax_kernel(const float* __restrict__ ref,
                        float* __restrict__ loc, float* __restrict__ attn)
{
    const long t = (long)blockIdx.x * blockDim.x + threadIdx.x;
    if (t >= (long)MROWS * NH_) return;
    const int  h  = (int)(t % NH_);
    const long bq = t / NH_;

    float* lp = loc  + bq * 256 + h * 32;   // (NL,NP,2)
    float* ap = attn + bq * 128 + h * 16;   // (NL*NP)
    const float* rp = ref + bq * (NL_ * 2); // (NL,2)

    // softmax over 16
    float v[16], mx = -1e30f;
#pragma unroll
    for (int i = 0; i < 16; ++i) { v[i] = ap[i]; mx = fmaxf(mx, v[i]); }
    float s = 0.f;
#pragma unroll
    for (int i = 0; i < 16; ++i) { v[i] = __expf(v[i] - mx); s += v[i]; }
    const float inv = 1.f / s;
#pragma unroll
    for (int i = 0; i < 16; ++i) ap[i] = v[i] * inv;

    // loc = ref + off / normalizer   (normalizer[l] = (W_l, H_l))
    const float nrm[NL_] = {128.f, 64.f, 32.f, 16.f}; // square levels: W_l==H_l
#pragma unroll
    for (int l = 0; l < NL_; ++l) {
        const float rx = rp[l * 2 + 0], ry = rp[l * 2 + 1];
        const float in = 1.f / nrm[l];
#pragma unroll
        for (int p = 0; p < NP_; ++p) {
            const int i = (l * NP_ + p) * 2;
            lp[i + 0] = rx + lp[i + 0] * in;
            lp[i + 1] = ry + lp[i + 1] * in;
        }
    }
}

// ---------------------------------------------------------------------------
// Deformable bilinear sampling core. One wave32 per (b,q,h); lane = channel d.
// value is (B, NK, NH*HD) row-major (GEMM output); fully L2-resident (~45MB).
// Scalar tap math is wave-uniform; per-tap reads are 128B contiguous lines.
// ---------------------------------------------------------------------------
__global__ __launch_bounds__(256)
void deform_core_kernel(const float* __restrict__ value,
                        const float* __restrict__ loc,
                        const float* __restrict__ attn,
                        float* __restrict__ core)
{
    const int  lane = threadIdx.x & 31;   // channel d in [0,32)
    const int  wave = threadIdx.x >> 5;
    const long item = (long)blockIdx.x * 8 + wave;
    if (item >= (long)MROWS * NH_) return;
    const int  h  = (int)(item % NH_);
    const long bq = item / NH_;
    const long b  = bq / NQ_;

    const float* lp = loc   + bq * 256 + h * 32;
    const float* ap = attn  + bq * 128 + h * 16;
    const float* vb = value + b * (long)NK_ * 256 + h * 32 + lane;

    const int HH[4] = {128, 64, 32, 16};
    const int SS[4] = {0, 16384, 20480, 21504};
    const int dxs[4] = {0, 1, 0, 1}, dys[4] = {0, 0, 1, 1};

    float acc = 0.f;
#pragma unroll
    for (int l = 0; l < 4; ++l) {
        const int   Hh = HH[l], Ww = HH[l], s = SS[l];
        const float fW = (float)Ww, fH = (float)Hh;
#pragma unroll
        for (int p = 0; p < 4; ++p) {
            const float aw  = ap[l * 4 + p];
            const float x   = lp[(l * 4 + p) * 2 + 0] * fW - 0.5f;
            const float y   = lp[(l * 4 + p) * 2 + 1] * fH - 0.5f;
            const float x0f = floorf(x), y0f = floorf(y);
            const int   x0  = (int)x0f,  y0  = (int)y0f;
            const float fx  = x - x0f,   fy  = y - y0f;
            const float wts[4] = {(1.f - fx) * (1.f - fy), fx * (1.f - fy),
                                  (1.f - fx) * fy,         fx * fy};
#pragma unroll
            for (int tp = 0; tp < 4; ++tp) {
                const int xi = x0 + dxs[tp], yi = y0 + dys[tp];
                if (xi >= 0 && xi < Ww && yi >= 0 && yi < Hh)
                    acc += aw * wts[tp] * vb[(long)(s + yi * Ww + xi) * 256];
            }
        }
    }
    core[bq * 256 + h * 32 + lane] = acc;
}

// ---------------------------------------------------------------------------
extern "C" void kernel_launch(void* const* d_in, const int* in_sizes, int n_in,
                              void* d_out, int out_size, void* d_ws, size_t ws_size,
                              hipStream_t stream)
{
    (void)in_sizes; (void)n_in; (void)out_size; (void)ws_size;

    const float* query  = (const float*)d_in[0];
    const float* ref    = (const float*)d_in[1];
    const float* inflat = (const float*)d_in[2];
    // d_in[3]/d_in[4]: spatial shapes / level starts — compile-time constants here
    const float* Wv     = (const float*)d_in[5];
    const float* bv     = (const float*)d_in[6];
    const float* Woff   = (const float*)d_in[7];
    const float* boff   = (const float*)d_in[8];
    const float* Wattn  = (const float*)d_in[9];
    const float* battn  = (const float*)d_in[10];
    const float* Wout   = (const float*)d_in[11];
    const float* bout   = (const float*)d_in[12];

    float* out      = (float*)d_out;                       // (B,NQ,256)
    float* out_loc  = out     + (long)MROWS * 256;         // (B,NQ,NH,NL,NP,2)
    float* out_attn = out_loc + (long)MROWS * 256;         // (B,NQ,NH,NL,NP)

    float* ws_value = (float*)d_ws;                        // (B,NK,256)
    float* ws_core  = ws_value + (long)MROWS * 256;        // (B,NQ,256)

    const dim3 blk(256);
    auto gblocks = [](int M, int N) {
        long t = (long)(M >> 6) * (N >> 4);   // 64-row x 16-col wave tiles
        return (unsigned)((t + 7) / 8);
    };

    // 1. value = input_flatten @ Wv + bv  -> workspace
    gemm256_wmma_f32<<<gblocks(MROWS, 256), blk, 0, stream>>>(inflat, Wv, bv, ws_value, MROWS, 256);
    // 2. off logits -> directly into d_out loc slot (same flat layout)
    gemm256_wmma_f32<<<gblocks(MROWS, 256), blk, 0, stream>>>(query, Woff, boff, out_loc, MROWS, 256);
    // 3. attn logits -> directly into d_out attn slot
    gemm256_wmma_f32<<<gblocks(MROWS, 128), blk, 0, stream>>>(query, Wattn, battn, out_attn, MROWS, 128);
    // 4. in-place: softmax(attn), loc = ref + off/normalizer
    loc_softmax_kernel<<<(unsigned)(((long)MROWS * NH_ + 255) / 256), blk, 0, stream>>>(ref, out_loc, out_attn);
    // 5. deformable sampling -> core (workspace)
    deform_core_kernel<<<(unsigned)(((long)MROWS * NH_ + 7) / 8), blk, 0, stream>>>(ws_value, out_loc, out_attn, ws_core);
    // 6. output = core @ Wout + bout -> d_out
    gemm256_wmma_f32<<<gblocks(MROWS, 256), blk, 0, stream>>>(ws_core, Wout, bout, out, MROWS, 256);
}